// HVQ_TR_switch_15745350108064
// MI455X (gfx1250) — compile-verified
//
#include <hip/hip_runtime.h>
#include <math.h>

// ---------------------------------------------------------------------------
// HVQ_TR_switch forward for MI455X (gfx1250, wave32, WMMA + TDM).
// B=512, L=196 (14x14), C=64, CH=128, NE=512, NX=15.
// Heavy math via V_WMMA_F32_16X16X4_F32; codebooks DMA'd into the 320KB LDS
// with TENSOR_LOAD_TO_LDS (hardware-inserted bank-deconflict padding).
// ---------------------------------------------------------------------------

typedef float v2f __attribute__((ext_vector_type(2)));
typedef float v8f __attribute__((ext_vector_type(8)));
typedef unsigned int u32x4 __attribute__((ext_vector_type(4)));
typedef int i32x4 __attribute__((ext_vector_type(4)));
typedef int i32x8 __attribute__((ext_vector_type(8)));

#define B_N   512
#define L_N   196
#define C_N   64
#define CH_N  128
#define NE_N  512
#define NX_N  15
#define MT_N  13          // ceil(196/16)

#define LDSP1 520         // LDS row stride (floats) for 64x512 codebook
#define LDSP2 520         // LDS row stride for 128x512 codebook
#define LDSP3 136         // LDS row stride for 128x128 W tile

// TDM pad codes: interval 0=2,1=4,...,7=256 DWORDs; amount 0=1,...,127=128 DWORDs
#define PADI_256 7        // pad after every 256 DWORDs
#define PADA_4   3        // by 4 DWORDs  -> 512-wide rows at stride 520 (+4 shift for col>=256)
#define PADI_128 6        // pad after every 128 DWORDs
#define PADA_8   7        // by 8 DWORDs  -> 128-wide rows at stride 136, no mid-row shift

// column shift for the 512-wide codebook layout (pad inserted mid-row at 256)
#define CSHIFT(c) ((c) + ((((c) >> 8) & 1) << 2))

// workspace layout (float offsets)
#define WS_P5   0                     // [512]  level-5 distance partials per block
#define WS_PHI  512                   // [2048] hi-level partials (lv*512+b)
#define WS_PF   2560                  // [512]  feature-loss partials per block
#define WS_E25  3072                  // [15*512]   |e|^2 for embed5
#define WS_E2HI 10752                 // [4*15*512] |e|^2 for embed_hi
#define WS_Q4   41472                 // [512*196*64] q4 (gathered level-5 codes)

// output layout (float offsets): rec, pred, loss, feature_loss, latent_loss, quant
#define OUT_REC   0LL
#define OUT_PRED  12845056LL
#define OUT_LOSS  12945408LL
#define OUT_FLOSS 12945409LL
#define OUT_LLOSS 12945410LL
#define OUT_QUANT 12945411LL

static __device__ __forceinline__ v8f wmma4(v2f a, v2f b, v8f c) {
  // D = A(16x4 f32) * B(4x16 f32) + C(16x16 f32)
  return __builtin_amdgcn_wmma_f32_16x16x4_f32(false, a, false, b, (short)0, c,
                                               false, false);
}

#if __has_builtin(__builtin_amdgcn_tensor_load_to_lds)
#define HAVE_TDM 1
// One-shot 2D tile DMA: rows x cols fp32 from global into LDS at ldsdst,
// with hardware padding (padi/pada codes) for bank de-conflict.
static __device__ __forceinline__ void tdm_load_2d(const float* gsrc, float* ldsdst,
                                                   int cols, int rows,
                                                   int padi, int pada) {
  unsigned long long ga = (unsigned long long)(uintptr_t)gsrc;
  u32x4 g0;
  g0.x = 1u;                                        // count=1 valid user descriptor
  g0.y = (unsigned)(uintptr_t)ldsdst;               // LDS byte offset (flat addr low 32)
  g0.z = (unsigned)(ga & 0xFFFFFFFFull);            // global_addr[31:0]
  g0.w = (unsigned)((ga >> 32) & 0x01FFFFFFull)     // global_addr[56:32]
       | 0x80000000u;                               // type=2 ("image")
  i32x8 g1;
  g1[0] = (2 << 16)                                 // data_size = 4 bytes
        | (1 << 20)                                 // pad_enable
        | (padi << 22) | (pada << 25);              // pad interval / amount
  g1[1] = (cols & 0xFFFF) << 16;                    // tensor_dim0[15:0]
  g1[2] = ((cols >> 16) & 0xFFFF)                   // tensor_dim0[31:16]
        | ((rows & 0xFFFF) << 16);                  // tensor_dim1[15:0]
  g1[3] = ((rows >> 16) & 0xFFFF)                   // tensor_dim1[31:16]
        | ((cols & 0xFFFF) << 16);                  // tile_dim0 = cols
  g1[4] = rows & 0xFFFF;                            // tile_dim1 = rows, tile_dim2 = 0
  g1[5] = cols;                                     // tensor_dim0_stride[31:0]
  g1[6] = 0;                                        // stride hi / dim1_stride lo
  g1[7] = 0;
  i32x4 gz = {0, 0, 0, 0};
#if __clang_major__ >= 23
  i32x8 gz8 = {0, 0, 0, 0, 0, 0, 0, 0};
  __builtin_amdgcn_tensor_load_to_lds(g0, g1, gz, gz, gz8, 0);
#else
  __builtin_amdgcn_tensor_load_to_lds(g0, g1, gz, gz, 0);
#endif
}
static __device__ __forceinline__ void tdm_wait() {
#if __has_builtin(__builtin_amdgcn_s_wait_tensorcnt)
  __builtin_amdgcn_s_wait_tensorcnt(0);
#endif
}
#endif

// ---------------------------------------------------------------------------
// Kernel 0: |e|^2 tables for all codebooks (tiny: 38400 dot products).
// ---------------------------------------------------------------------------
__global__ void k_e2(const float* __restrict__ e5, const float* __restrict__ ehi,
                     float* __restrict__ ws) {
  int t = blockIdx.x * 256 + threadIdx.x;
  if (t < NX_N * NE_N) {
    int cb = t / NE_N, n = t % NE_N;
    const float* eb = e5 + (size_t)cb * C_N * NE_N;
    float s = 0.f;
    for (int d = 0; d < C_N; ++d) { float v = eb[(size_t)d * NE_N + n]; s += v * v; }
    ws[WS_E25 + t] = s;
  }
  int t2 = t - NX_N * NE_N;
  if (t2 >= 0 && t2 < 4 * NX_N * NE_N) {
    int lv = t2 / (NX_N * NE_N);
    int r  = t2 % (NX_N * NE_N);
    int cb = r / NE_N, n = r % NE_N;
    const float* eb = ehi + ((size_t)(lv * NX_N + cb) * (2 * C_N)) * NE_N;
    float s = 0.f;
    for (int d = 0; d < 2 * C_N; ++d) { float v = eb[(size_t)d * NE_N + n]; s += v * v; }
    ws[WS_E2HI + t2] = s;
  }
}

// ---------------------------------------------------------------------------
// Kernel 1: level-5 VQ. One block (4 waves) per sample b.
// dist(m,n) = x2[m] + (e2[n] - 2*x.e); argmin over n; min value == diff term.
// ---------------------------------------------------------------------------
__global__ void __launch_bounds__(128)
k_vq5(const float* __restrict__ enc, const float* __restrict__ e5,
      const int* __restrict__ label, float* __restrict__ ws) {
  extern __shared__ float smem[];
  float* elds   = smem;                    // [64][LDSP1] codebook
  float* s_e2   = elds + 64 * LDSP1;       // [512]
  float* s_part = s_e2 + 512;              // [4]

  const int b = blockIdx.x;
  const int tid = threadIdx.x, lane = tid & 31, w = tid >> 5;
  const int ln = lane & 15, hi = lane >> 4;
  const int cb = label[b];

  // stage codebook (64x512, 128KB) into padded LDS
  const float* eb = e5 + (size_t)cb * C_N * NE_N;
#ifdef HAVE_TDM
  if (tid == 0) tdm_load_2d(eb, elds, NE_N, C_N, PADI_256, PADA_4);
#else
  for (int i = tid; i < (C_N * NE_N) / 4; i += 128) {
    int r = i >> 7, c = (i & 127) * 4;
    *(float4*)(elds + r * LDSP1 + CSHIFT(c)) = ((const float4*)eb)[i];
  }
#endif
  for (int i = tid; i < NE_N; i += 128) s_e2[i] = ws[WS_E25 + cb * NE_N + i];
#ifdef HAVE_TDM
  tdm_wait();
#endif
  __syncthreads();

  const float* xb = enc + ((size_t)4 * B_N + b) * (L_N * C_N);
  float* q4 = ws + WS_Q4 + (size_t)b * L_N * C_N;
  float diffacc = 0.f;

  for (int mt = w; mt < MT_N; mt += 4) {
    const int Lb = mt * 16;
    int row = Lb + ln; if (row > L_N - 1) row = L_N - 1;   // clamp pad rows
    const float* xr = xb + (size_t)row * C_N;

    v2f a[16];
    float x2p = 0.f;
#pragma unroll
    for (int ks = 0; ks < 16; ++ks) {
      a[ks] = *(const v2f*)(xr + 4 * ks + 2 * hi);
      x2p += a[ks].x * a[ks].x + a[ks].y * a[ks].y;
    }
    float x2m = x2p + __shfl_xor(x2p, 16, 32);             // full row |x|^2

    float bd[8]; int bi[8];
#pragma unroll
    for (int r = 0; r < 8; ++r) { bd[r] = 3.4e38f; bi[r] = 0; }

    for (int nt = 0; nt < NE_N / 16; ++nt) {
      const int nb = nt * 16;
      const int nc = CSHIFT(nb) + ln;      // padded column (uniform shift per tile)
      const float e2v = s_e2[nb + ln];
      v8f acc = {};
#pragma unroll
      for (int ks = 0; ks < 16; ++ks) {
        const int kr = 4 * ks + 2 * hi;
        v2f bf = { elds[kr * LDSP1 + nc], elds[(kr + 1) * LDSP1 + nc] };
        acc = wmma4(a[ks], bf, acc);
      }
#pragma unroll
      for (int r = 0; r < 8; ++r) {
        float dv = e2v - 2.f * acc[r];
        if (dv < bd[r]) { bd[r] = dv; bi[r] = nb + ln; }
      }
    }

    // butterfly min+argmin across the 16 lanes of each half (rows r+8*hi)
#pragma unroll
    for (int r = 0; r < 8; ++r) {
#pragma unroll
      for (int mask = 1; mask <= 8; mask <<= 1) {
        float od = __shfl_xor(bd[r], mask, 32);
        int   oi = __shfl_xor(bi[r], mask, 32);
        if (od < bd[r] || (od == bd[r] && oi < bi[r])) { bd[r] = od; bi[r] = oi; }
      }
      const int  rrow   = r + 8 * hi;
      const bool writer = (ln == (hi ? r + 8 : r));        // lane whose x2m == x2[rrow]
      if (writer && (Lb + rrow) < L_N) diffacc += x2m + bd[r];
    }

    // gather winning codes -> q4 (column id of codebook)
#pragma unroll
    for (int rowi = 0; rowi < 16; ++rowi) {
      int id = __shfl(bi[rowi & 7], (rowi < 8) ? 0 : 16, 32);
      if (Lb + rowi < L_N) {
        int idc = CSHIFT(id);
        float* qr = q4 + (size_t)(Lb + rowi) * C_N;
        qr[lane]      = elds[lane * LDSP1 + idc];
        qr[lane + 32] = elds[(lane + 32) * LDSP1 + idc];
      }
    }
  }

#pragma unroll
  for (int mask = 1; mask <= 16; mask <<= 1) diffacc += __shfl_xor(diffacc, mask, 32);
  if (lane == 0) s_part[w] = diffacc;
  __syncthreads();
  if (tid == 0) ws[WS_P5 + b] = s_part[0] + s_part[1] + s_part[2] + s_part[3];
}

// ---------------------------------------------------------------------------
// Kernel 2: hi-level VQ (4 levels). Block = (b, level). x = concat(enc_src, q4).
// Gathered codes are written straight to the quant_stack output.
// ---------------------------------------------------------------------------
__global__ void __launch_bounds__(128)
k_vqhi(const float* __restrict__ enc, const float* __restrict__ ehi,
       const int* __restrict__ label, float* __restrict__ ws,
       float* __restrict__ out) {
  extern __shared__ float smem[];
  float* elds   = smem;                    // [128][LDSP2] codebook (256KB < 320KB WGP LDS)
  float* s_e2   = elds + 128 * LDSP2;      // [512]
  float* s_part = s_e2 + 512;              // [4]

  const int b = blockIdx.x, lv = blockIdx.y;
  const int tid = threadIdx.x, lane = tid & 31, w = tid >> 5;
  const int ln = lane & 15, hi = lane >> 4;
  const int cb = label[b];
  const int src = 3 - lv;                  // levels use enc[3],enc[2],enc[1],enc[0]

  const float* eb = ehi + ((size_t)(lv * NX_N + cb) * (2 * C_N)) * NE_N;
#ifdef HAVE_TDM
  if (tid == 0) tdm_load_2d(eb, elds, NE_N, 2 * C_N, PADI_256, PADA_4);
#else
  for (int i = tid; i < (2 * C_N * NE_N) / 4; i += 128) {
    int r = i >> 7, c = (i & 127) * 4;
    *(float4*)(elds + r * LDSP2 + CSHIFT(c)) = ((const float4*)eb)[i];
  }
#endif
  for (int i = tid; i < NE_N; i += 128)
    s_e2[i] = ws[WS_E2HI + (lv * NX_N + cb) * NE_N + i];
#ifdef HAVE_TDM
  tdm_wait();
#endif
  __syncthreads();

  const float* xb  = enc + ((size_t)src * B_N + b) * (L_N * C_N);
  const float* x2b = ws + WS_Q4 + (size_t)b * L_N * C_N;
  float* qo = out + OUT_QUANT + ((size_t)lv * B_N + b) * (size_t)L_N * (2 * C_N);
  float diffacc = 0.f;

  for (int mt = w; mt < MT_N; mt += 4) {
    const int Lb = mt * 16;
    int row = Lb + ln; if (row > L_N - 1) row = L_N - 1;
    const float* xr  = xb  + (size_t)row * C_N;
    const float* x2r = x2b + (size_t)row * C_N;

    v2f a[32];
    float x2p = 0.f;
#pragma unroll
    for (int ks = 0; ks < 16; ++ks) {
      a[ks] = *(const v2f*)(xr + 4 * ks + 2 * hi);
      x2p += a[ks].x * a[ks].x + a[ks].y * a[ks].y;
    }
#pragma unroll
    for (int ks = 0; ks < 16; ++ks) {
      a[16 + ks] = *(const v2f*)(x2r + 4 * ks + 2 * hi);
      x2p += a[16 + ks].x * a[16 + ks].x + a[16 + ks].y * a[16 + ks].y;
    }
    float x2m = x2p + __shfl_xor(x2p, 16, 32);

    float bd[8]; int bi[8];
#pragma unroll
    for (int r = 0; r < 8; ++r) { bd[r] = 3.4e38f; bi[r] = 0; }

    for (int nt = 0; nt < NE_N / 16; ++nt) {
      const int nb = nt * 16;
      const int nc = CSHIFT(nb) + ln;
      const float e2v = s_e2[nb + ln];
      v8f acc = {};
#pragma unroll
      for (int ks = 0; ks < 32; ++ks) {
        const int kr = 4 * ks + 2 * hi;
        v2f bf = { elds[kr * LDSP2 + nc], elds[(kr + 1) * LDSP2 + nc] };
        acc = wmma4(a[ks], bf, acc);
      }
#pragma unroll
      for (int r = 0; r < 8; ++r) {
        float dv = e2v - 2.f * acc[r];
        if (dv < bd[r]) { bd[r] = dv; bi[r] = nb + ln; }
      }
    }

#pragma unroll
    for (int r = 0; r < 8; ++r) {
#pragma unroll
      for (int mask = 1; mask <= 8; mask <<= 1) {
        float od = __shfl_xor(bd[r], mask, 32);
        int   oi = __shfl_xor(bi[r], mask, 32);
        if (od < bd[r] || (od == bd[r] && oi < bi[r])) { bd[r] = od; bi[r] = oi; }
      }
      const int  rrow   = r + 8 * hi;
      const bool writer = (ln == (hi ? r + 8 : r));
      if (writer && (Lb + rrow) < L_N) diffacc += x2m + bd[r];
    }

#pragma unroll
    for (int rowi = 0; rowi < 16; ++rowi) {
      int id = __shfl(bi[rowi & 7], (rowi < 8) ? 0 : 16, 32);
      if (Lb + rowi < L_N) {
        int idc = CSHIFT(id);
        float* qr = qo + (size_t)(Lb + rowi) * (2 * C_N);
#pragma unroll
        for (int j = 0; j < 4; ++j)
          qr[lane + 32 * j] = elds[(lane + 32 * j) * LDSP2 + idc];
      }
    }
  }

#pragma unroll
  for (int mask = 1; mask <= 16; mask <<= 1) diffacc += __shfl_xor(diffacc, mask, 32);
  if (lane == 0) s_part[w] = diffacc;
  __syncthreads();
  if (tid == 0) ws[WS_PHI + lv * B_N + b] = s_part[0] + s_part[1] + s_part[2] + s_part[3];
}

// ---------------------------------------------------------------------------
// Kernel 3: rec = dec_b x W[label_b] + bias; pred = sqrt(sum_ch (rec-org)^2);
// feature_loss partials. One block per sample.
// ---------------------------------------------------------------------------
__global__ void __launch_bounds__(128)
k_rec(const float* __restrict__ dec, const float* __restrict__ org,
      const float* __restrict__ Wo, const float* __restrict__ bo,
      const int* __restrict__ label, float* __restrict__ ws,
      float* __restrict__ out) {
  extern __shared__ float smem[];
  float* wlds   = smem;                    // [128][LDSP3]
  float* s_bias = wlds + 128 * LDSP3;      // [128]
  float* s_part = s_bias + 128;            // [4]

  const int b = blockIdx.x;
  const int tid = threadIdx.x, lane = tid & 31, w = tid >> 5;
  const int ln = lane & 15, hi = lane >> 4;
  const int cb = label[b];

  const float* wb = Wo + (size_t)cb * (2 * C_N) * CH_N;
#ifdef HAVE_TDM
  if (tid == 0) tdm_load_2d(wb, wlds, CH_N, 2 * C_N, PADI_128, PADA_8);
#else
  for (int i = tid; i < (2 * C_N * CH_N) / 4; i += 128) {
    int r = i >> 5, c4 = i & 31;                     // 32 float4 per 128-wide row
    *(float4*)(wlds + r * LDSP3 + c4 * 4) = ((const float4*)wb)[i];
  }
#endif
  if (tid < CH_N) s_bias[tid] = bo[cb * CH_N + tid];
#ifdef HAVE_TDM
  tdm_wait();
#endif
  __syncthreads();

  float flacc = 0.f;

  for (int mt = w; mt < MT_N; mt += 4) {
    const int Lb = mt * 16;
    int row = Lb + ln; if (row > L_N - 1) row = L_N - 1;
    const float* dr = dec + ((size_t)row * B_N + b) * (2 * C_N);

    v2f a[32];
#pragma unroll
    for (int ks = 0; ks < 32; ++ks) a[ks] = *(const v2f*)(dr + 4 * ks + 2 * hi);

    float pacc[8];
#pragma unroll
    for (int r = 0; r < 8; ++r) pacc[r] = 0.f;

    for (int nt = 0; nt < CH_N / 16; ++nt) {
      const int nb = nt * 16;
      v8f acc = {};
#pragma unroll
      for (int ks = 0; ks < 32; ++ks) {
        const int kr = 4 * ks + 2 * hi;
        v2f bf = { wlds[kr * LDSP3 + nb + ln], wlds[(kr + 1) * LDSP3 + nb + ln] };
        acc = wmma4(a[ks], bf, acc);
      }
      const float bias = s_bias[nb + ln];
      float rv[8];
#pragma unroll
      for (int r = 0; r < 8; ++r) rv[r] = acc[r] + bias;

      const size_t obase = (size_t)b * CH_N * L_N + (size_t)(nb + ln) * L_N + Lb;
#pragma unroll
      for (int r = 0; r < 8; ++r) {
        const int m = r + 8 * hi;
        if (Lb + m < L_N) {
          float dv = rv[r] - org[obase + m];
          float d2 = dv * dv;
          pacc[r] += d2;
          flacc   += d2;
        }
      }
      // rec stores: each lane writes 2 contiguous float4 (global_store_b128)
      float* ob = out + OUT_REC + obase;
      float4 lo4 = make_float4(rv[0], rv[1], rv[2], rv[3]);   // rows 8*hi .. +3
      float4 hi4 = make_float4(rv[4], rv[5], rv[6], rv[7]);   // rows 8*hi+4 .. +7
      if (Lb + 8 * hi + 3 < L_N) *(float4*)(ob + 8 * hi)     = lo4;
      if (Lb + 8 * hi + 7 < L_N) *(float4*)(ob + 8 * hi + 4) = hi4;
    }

    // per-row channel SSQ -> pred
#pragma unroll
    for (int r = 0; r < 8; ++r) {
#pragma unroll
      for (int mask = 1; mask <= 8; mask <<= 1) pacc[r] += __shfl_xor(pacc[r], mask, 32);
      const int  m      = r + 8 * hi;
      const bool writer = (ln == (hi ? r + 8 : r));
      if (writer && (Lb + m) < L_N)
        out[OUT_PRED + (size_t)b * L_N + Lb + m] = sqrtf(pacc[r]);
    }
  }

#pragma unroll
  for (int mask = 1; mask <= 16; mask <<= 1) flacc += __shfl_xor(flacc, mask, 32);
  if (lane == 0) s_part[w] = flacc;
  __syncthreads();
  if (tid == 0) ws[WS_PF + b] = s_part[0] + s_part[1] + s_part[2] + s_part[3];
}

// ---------------------------------------------------------------------------
// Kernel 4: deterministic final reduction of the loss scalars.
// ---------------------------------------------------------------------------
__global__ void k_final(const float* __restrict__ ws, float* __restrict__ out) {
  if (blockIdx.x == 0 && threadIdx.x == 0) {
    float s5 = 0.f, sh = 0.f, sf = 0.f;
    for (int i = 0; i < B_N; ++i)     s5 += ws[WS_P5 + i];
    for (int i = 0; i < 4 * B_N; ++i) sh += ws[WS_PHI + i];
    for (int i = 0; i < B_N; ++i)     sf += ws[WS_PF + i];
    float latent = s5 / (float)(B_N * L_N * C_N) + sh / (float)(B_N * L_N * 2 * C_N);
    float fl     = sf / (float)(B_N * CH_N * L_N);
    out[OUT_LLOSS] = latent;
    out[OUT_FLOSS] = fl;
    out[OUT_LOSS]  = 0.25f * latent + fl;
  }
}

// ---------------------------------------------------------------------------
extern "C" void kernel_launch(void* const* d_in, const int* in_sizes, int n_in,
                              void* d_out, int out_size, void* d_ws, size_t ws_size,
                              hipStream_t stream) {
  const float* enc = (const float*)d_in[0];   // [5,512,196,64]
  const float* dec = (const float*)d_in[1];   // [196,512,128]
  const float* org = (const float*)d_in[2];   // [512,128,14,14]
  const float* e5  = (const float*)d_in[3];   // [15,64,512]
  const float* ehi = (const float*)d_in[4];   // [4,15,128,512]
  const float* Wo  = (const float*)d_in[5];   // [15,128,128]
  const float* bo  = (const float*)d_in[6];   // [15,128]
  const int*   lab = (const int*)d_in[7];     // [512]
  float* out = (float*)d_out;
  float* ws  = (float*)d_ws;

  const size_t sh1 = (size_t)(64 * LDSP1 + 512 + 8) * sizeof(float);   // ~135 KB
  const size_t sh2 = (size_t)(128 * LDSP2 + 512 + 8) * sizeof(float);  // ~268 KB
  const size_t sh3 = (size_t)(128 * LDSP3 + 128 + 8) * sizeof(float);  // ~70 KB
  (void)hipFuncSetAttribute((const void*)k_vq5,  hipFuncAttributeMaxDynamicSharedMemorySize, (int)sh1);
  (void)hipFuncSetAttribute((const void*)k_vqhi, hipFuncAttributeMaxDynamicSharedMemorySize, (int)sh2);
  (void)hipFuncSetAttribute((const void*)k_rec,  hipFuncAttributeMaxDynamicSharedMemorySize, (int)sh3);

  k_e2<<<150, 256, 0, stream>>>(e5, ehi, ws);
  k_vq5<<<B_N, 128, sh1, stream>>>(enc, e5, lab, ws);
  k_vqhi<<<dim3(B_N, 4), 128, sh2, stream>>>(enc, ehi, lab, ws, out);
  k_rec<<<B_N, 128, sh3, stream>>>(dec, org, Wo, bo, lab, ws, out);
  k_final<<<1, 1, 0, stream>>>(ws, out);
}